// CTCLOSS_71090298683644
// MI455X (gfx1250) — compile-verified
//
#include <hip/hip_runtime.h>
#include <cstdint>
#include <cstddef>

#define DEVFN __device__ __forceinline__

typedef float v2f __attribute__((ext_vector_type(2)));
typedef float v8f __attribute__((ext_vector_type(8)));
typedef int   v4i __attribute__((vector_size(16)));   // matches builtin's param type
typedef __attribute__((address_space(1))) v4i* gas_b128_t;
typedef __attribute__((address_space(3))) v4i* las_b128_t;

constexpr int   kT  = 1024;          // timesteps
constexpr int   kN  = 128;           // batch
constexpr int   kC  = 32;            // classes (incl. blank=0)
constexpr int   kS  = 128;           // target length
constexpr int   kL  = 2 * kS + 1;    // 257 extended-lattice states
constexpr int   kLP = 272;           // padded to 17 WMMA tiles of 16
constexpr int   kNT = kLP / 16;      // 17 s-tiles
constexpr int   kTB = 16;            // timesteps per staged block
constexpr int   kNB = kT / kTB;      // 64 blocks
constexpr float kNegInf = -1e30f;

#if __has_builtin(__builtin_amdgcn_wmma_f32_16x16x4_f32)
#define CTC_WMMA 1
#else
#define CTC_WMMA 0
#endif

#if __has_builtin(__builtin_amdgcn_global_load_async_to_lds_b128) && \
    __has_builtin(__builtin_amdgcn_s_wait_asynccnt)
#define CTC_ASYNC 1
#else
#define CTC_ASYNC 0
#endif

DEVFN float lae(float a, float b) {
  // logaddexp; inputs are finite (NEG_INF = -1e30 stand-in), so no NaN paths.
  float m = fmaxf(a, b);
  float d = fabsf(a - b);
  return m + log1pf(expf(-d));
}

// Stage one 16x32 block of log_probs for sample n into LDS (row-major [t'][c]).
DEVFN void stage_block(const float* __restrict__ lp, int n, int t0,
                       float* dst, int lane) {
#if CTC_ASYNC
  // 4 async b128 ops move the whole 2KB block: lane l -> row l/8, 16B chunk l%8.
  const int r = lane >> 3;
  const int q = lane & 7;
#pragma unroll
  for (int c4 = 0; c4 < 4; ++c4) {
    const int tp = c4 * 4 + r;
    const float* g = lp + ((size_t)(t0 + tp) * kN + n) * kC + q * 4;
    float* l = dst + tp * kC + q * 4;
    __builtin_amdgcn_global_load_async_to_lds_b128(
        (gas_b128_t)g, (las_b128_t)l, 0, 0);
  }
#else
#pragma unroll
  for (int tp = 0; tp < kTB; ++tp)
    dst[tp * kC + lane] = lp[((size_t)(t0 + tp) * kN + n) * kC + lane];
#endif
}

DEVFN void wait_stage(bool one_block_pending) {
#if CTC_ASYNC
  if (one_block_pending) __builtin_amdgcn_s_wait_asynccnt(4);  // 4 newest may remain
  else                   __builtin_amdgcn_s_wait_asynccnt(0);
#else
  (void)one_block_pending;
#endif
}

__global__ __launch_bounds__(32) void ctc_alpha_kernel(
    const float* __restrict__ lp,       // (T,N,C) log-softmax
    const int*   __restrict__ targets,  // (N,S)
    const int*   __restrict__ sizes,    // (N,)
    const int*   __restrict__ tsizes,   // (N,)
    float*       __restrict__ loss)     // (N,) per-sample loss/target_len
{
  __shared__ __align__(16) float pb[2][kTB * kC];  // staged log-prob blocks
  __shared__ float lpext[kTB * kLP];               // gathered lp_ext for a block
  __shared__ float alpha[2][kLP + 2];              // double buffer, 2 pad slots
  __shared__ int   ext_s[kLP];
  __shared__ int   skip_s[kLP];

  const int n    = blockIdx.x;
  const int lane = threadIdx.x;  // wave32: one wave per workgroup
  const int len  = sizes[n];
  const int tl   = tsizes[n];

  // Kick off staging of block 0 ASAP; it overlaps the lattice setup below.
  stage_block(lp, n, 0, pb[0], lane);

  // Extended labels: ext[2i]=blank(0), ext[2i+1]=targets[i]; skip mask per ref.
  for (int j = 0; j < 9; ++j) {
    int s = j * 32 + lane;
    if (s < kLP) {
      int e = 0, sk = 0;
      if (s & 1) {
        int i = (s - 1) >> 1;
        if (i < kS) {
          e = targets[n * kS + i];
          if (s >= 3) sk = (targets[n * kS + i] != targets[n * kS + i - 1]);
        }
      }
      ext_s[s]  = e;   // label is never blank on odd s, so skip needs no !=0 test
      skip_s[s] = sk;
    }
  }
  if (lane < 2) { alpha[0][lane] = kNegInf; alpha[1][lane] = kNegInf; }  // pads
  __builtin_amdgcn_wave_barrier();

#if CTC_WMMA
  const int m = lane & 15;   // A: M row / B,D: N column handled by this lane
  const int h = lane >> 4;   // lane half selects K (A/B) or M+8 (D)
  // ext is time-invariant: hoist each tile's column label into registers so the
  // WMMA phase has zero DS loads (no s_wait_dscnt between tile chains).
  int extv[kNT];
#pragma unroll
  for (int tile = 0; tile < kNT; ++tile) extv[tile] = ext_s[tile * 16 + m];
#endif
  // skip mask is also time-invariant: keep this lane's 9 states in registers.
  int skipv[9];
#pragma unroll
  for (int j = 0; j < 9; ++j) {
    int s = j * 32 + lane;
    skipv[j] = (s < kLP) ? skip_s[s] : 0;
  }

  int p = 0;  // alpha parity
  for (int b = 0; b < kNB; ++b) {
    const int t0 = b * kTB;
    if (b + 1 < kNB) { stage_block(lp, n, t0 + kTB, pb[(b + 1) & 1], lane); wait_stage(true); }
    else             { wait_stage(false); }
    __builtin_amdgcn_wave_barrier();
    asm volatile("" ::: "memory");

    const float* P = pb[b & 1];

    // ---- lp_ext[t'][s] = P[t'][ext[s]] via one-hot matmul on the matrix pipe ----
#if CTC_WMMA
    v2f aop[8];
#pragma unroll
    for (int kk = 0; kk < 8; ++kk) {          // A: 16x4 f32 tiles, K = kk*4..kk*4+3
      const int c0 = kk * 4 + 2 * h;          // lane l, vgpr v -> K = 2*(l/16)+v
      aop[kk].x = P[m * kC + c0];
      aop[kk].y = P[m * kC + c0 + 1];
    }
#pragma unroll
    for (int tile = 0; tile < kNT; ++tile) {  // 17 column tiles of s
      const int s0 = tile * 16;
      const int e  = extv[tile];
      v8f acc = {0.f, 0.f, 0.f, 0.f, 0.f, 0.f, 0.f, 0.f};
#pragma unroll
      for (int kk = 0; kk < 8; ++kk) {
        const int kb = kk * 4 + 2 * h;        // B: one-hot E[k][s] = (ext[s]==k)
        v2f bop;
        bop.x = (e == kb)     ? 1.0f : 0.0f;
        bop.y = (e == kb + 1) ? 1.0f : 0.0f;
        acc = __builtin_amdgcn_wmma_f32_16x16x4_f32(
            false, aop[kk], false, bop, (short)0, acc, false, false);
      }
#pragma unroll
      for (int r = 0; r < 8; ++r)             // D: lane l, vgpr r -> M = r + 8*(l/16)
        lpext[(h * 8 + r) * kLP + s0 + m] = acc[r];
    }
#else
    for (int tp = 0; tp < kTB; ++tp)
      for (int j = 0; j < 9; ++j) {
        int s = j * 32 + lane;
        if (s < kLP) lpext[tp * kLP + s] = P[tp * kC + ext_s[s]];
      }
#endif
    __builtin_amdgcn_wave_barrier();

    // ---- alpha recursion (wave-synchronous; DS ops are in-order per wave) ----
    for (int tp = 0; tp < kTB; ++tp) {
      const int t = t0 + tp;
      float* ap = &alpha[p][2];
      float* an = &alpha[p ^ 1][2];
      if (t == 0) {
        for (int j = 0; j < 9; ++j) {
          int s = j * 32 + lane;
          if (s < kL) ap[s] = (s < 2) ? lpext[s] : kNegInf;
        }
        __builtin_amdgcn_wave_barrier();
        if (len == 1 && lane == 0)
          loss[n] = -lae(ap[2 * tl], ap[2 * tl - 1]) / (float)tl;
        continue;
      }
#pragma unroll
      for (int j = 0; j < 9; ++j) {
        int s = j * 32 + lane;
        if (s < kL) {
          float a0 = ap[s];
          float a1 = ap[s - 1];                           // pad slot covers s==0
          float a2 = skipv[j] ? ap[s - 2] : kNegInf;      // pads cover s<2
          an[s] = lae(lae(a0, a1), a2) + lpext[tp * kLP + s];
        }
      }
      __builtin_amdgcn_wave_barrier();
      if (t == len - 1 && lane == 0)
        loss[n] = -lae(an[2 * tl], an[2 * tl - 1]) / (float)tl;
      p ^= 1;
    }
  }
}

__global__ __launch_bounds__(128) void ctc_reduce_kernel(
    const float* __restrict__ loss, float* __restrict__ out) {
  __shared__ float r[kN];
  const int i = threadIdx.x;
  r[i] = loss[i];
  __syncthreads();
  for (int s = kN / 2; s > 0; s >>= 1) {
    if (i < s) r[i] += r[i + s];
    __syncthreads();
  }
  if (i == 0) out[0] = r[0] / (float)kN;   // fixed tree -> deterministic
}

extern "C" void kernel_launch(void* const* d_in, const int* in_sizes, int n_in,
                              void* d_out, int out_size, void* d_ws, size_t ws_size,
                              hipStream_t stream) {
  (void)in_sizes; (void)n_in; (void)out_size; (void)ws_size;
  const float* lp   = (const float*)d_in[0];
  const int*   tgt  = (const int*)d_in[1];
  const int*   szs  = (const int*)d_in[2];
  const int*   tsz  = (const int*)d_in[3];
  float*       ws   = (float*)d_ws;     // kN per-sample losses
  float*       out  = (float*)d_out;

  ctc_alpha_kernel<<<dim3(kN), dim3(32), 0, stream>>>(lp, tgt, szs, tsz, ws);
  ctc_reduce_kernel<<<dim3(1), dim3(kN), 0, stream>>>(ws, out);
}